// GaussianBlur_70231305224809
// MI455X (gfx1250) — compile-verified
//
#include <hip/hip_runtime.h>
#include <stdint.h>

// Gaussian blur 7x7 (sigma=1.5), reflect-101, NHWC 64x512x512x3 fp32.
// Fused separable passes through LDS; async global->LDS tile loads (CDNA5).

#define KS     7
#define PAD    3
#define W      512
#define H      512
#define C      3
#define ROWS   8                       // output rows per block
#define ROWF   (W * C)                 // 1536 floats per image row
#define ROW4   (ROWF / 4)              // 384 float4 per row
#define VROWF  (ROWF + 2 * PAD * C)    // 1554: vertical-pass row with horiz halo
#define INROWS (ROWS + KS - 1)         // 14 input rows per tile
#define NTHR   256                     // 8 wave32s

typedef float v4f __attribute__((ext_vector_type(4)));
typedef int   v4i __attribute__((ext_vector_type(4)));
typedef __attribute__((address_space(1))) v4i GlobV4; // global int4
typedef __attribute__((address_space(3))) v4i LdsV4;  // LDS int4

#if defined(__HIP_DEVICE_COMPILE__) && \
    __has_builtin(__builtin_amdgcn_global_load_async_to_lds_b128)
#define USE_ASYNC_LDS 1
#else
#define USE_ASYNC_LDS 0
#endif

__device__ __forceinline__ void wait_async_zero() {
#if __has_builtin(__builtin_amdgcn_s_wait_asynccnt)
  __builtin_amdgcn_s_wait_asynccnt(0);
#else
  asm volatile("s_wait_asynccnt 0" ::: "memory");
#endif
}

__global__ __launch_bounds__(NTHR) void gauss7_fused(
    const float* __restrict__ src, float* __restrict__ dst) {
  extern __shared__ float smem[];
  float* intile = smem;                  // INROWS * ROWF  (raw input strip)
  float* vtile  = smem + INROWS * ROWF;  // ROWS * VROWF   (vertical-pass result)

  const int tid = threadIdx.x;
  const int r0  = blockIdx.x * ROWS;
  const long long img = (long long)blockIdx.y * (long long)(H * ROWF);

  const float wk[KS] = {0.03663285f, 0.11128076f, 0.21674533f, 0.27068216f,
                        0.21674533f, 0.11128076f, 0.03663285f};

  // ---- Stage 1: global -> LDS, 16B per lane, reflect-101 row remap ----
  for (int i = tid; i < INROWS * ROW4; i += NTHR) {
    int row = i / ROW4;
    int x4  = i - row * ROW4;
    int ir  = r0 - PAD + row;
    ir = (ir < 0) ? -ir : ir;                    // reflect101 top
    ir = (ir > H - 1) ? (2 * (H - 1) - ir) : ir; // reflect101 bottom
    const float* gp = src + img + (long long)ir * ROWF + x4 * 4;
    float* lp = intile + row * ROWF + x4 * 4;
#if USE_ASYNC_LDS
    __builtin_amdgcn_global_load_async_to_lds_b128((GlobV4*)gp, (LdsV4*)lp, 0, 0);
#else
    *(v4f*)lp = *(const v4f*)gp;
#endif
  }
#if USE_ASYNC_LDS
  wait_async_zero();
#endif
  __syncthreads();

  // ---- Stage 2: vertical 7-tap, LDS -> LDS (interior of padded vtile) ----
  for (int i = tid; i < ROWS * ROWF; i += NTHR) {
    int row = i / ROWF;
    int xx  = i - row * ROWF;
    const float* cp = intile + row * ROWF + xx;
    float acc = wk[0] * cp[0];
#pragma unroll
    for (int k = 1; k < KS; ++k) acc += wk[k] * cp[k * ROWF];
    vtile[row * VROWF + PAD * C + xx] = acc;
  }
  __syncthreads();

  // Fill reflect-101 horizontal halo (9 floats each side per row)
  if (tid < ROWS * 2 * PAD * C) {  // 144 lanes
    int row = tid / (2 * PAD * C);
    int j   = tid - row * (2 * PAD * C);
    float* vr = vtile + row * VROWF;
    if (j < PAD * C) {                         // left: pixel -3+j/C -> reflect
      int p = PAD - j / C;                     // 3,2,1
      vr[j] = vr[PAD * C + p * C + (j % C)];
    } else {                                   // right: pixel 512+jj/C -> 510-jj/C
      int jj = j - PAD * C;
      int p  = (W - 2) - jj / C;               // 510,509,508
      vr[PAD * C + ROWF + jj] = vr[PAD * C + p * C + (jj % C)];
    }
  }
  __syncthreads();

  // ---- Stage 3: horizontal 7-tap, LDS -> registers -> NT float4 store ----
  const long long obase = img + (long long)r0 * ROWF;
  for (int i = tid; i < ROWS * ROW4; i += NTHR) {
    int row = i / ROW4;
    int x4  = i - row * ROW4;
    const float* vr = vtile + row * VROWF + PAD * C + x4 * 4;
    v4f o;
#pragma unroll
    for (int e = 0; e < 4; ++e) {
      float acc = wk[0] * vr[e - PAD * C];
#pragma unroll
      for (int k = 1; k < KS; ++k) acc += wk[k] * vr[e + (k - PAD) * C];
      o[e] = acc;
    }
    __builtin_nontemporal_store(
        o, (v4f*)(dst + obase + (long long)row * ROWF + x4 * 4));
  }
}

extern "C" void kernel_launch(void* const* d_in, const int* in_sizes, int n_in,
                              void* d_out, int out_size, void* d_ws, size_t ws_size,
                              hipStream_t stream) {
  const float* x = (const float*)d_in[0];
  float* out = (float*)d_out;
  (void)d_ws; (void)ws_size; (void)n_in; (void)out_size;

  const int N = in_sizes[0] / (H * ROWF);  // 64 images
  dim3 grid(H / ROWS, N, 1);               // (64, 64)
  size_t smem = (size_t)(INROWS * ROWF + ROWS * VROWF) * sizeof(float); // 135744 B
  gauss7_fused<<<grid, NTHR, smem, stream>>>(x, out);
}